// gcn_89541478187049
// MI455X (gfx1250) — compile-verified
//
#include <hip/hip_runtime.h>

typedef float v2f __attribute__((ext_vector_type(2)));
typedef float v8f __attribute__((ext_vector_type(8)));

#define K_DIM 128

#if defined(__HIP_DEVICE_COMPILE__) && __has_builtin(__builtin_amdgcn_global_load_async_to_lds_b128)
#define HAVE_ASYNC_LDS 1
typedef int v4i_t __attribute__((vector_size(4 * sizeof(int))));
typedef __attribute__((address_space(1))) v4i_t g_v4i;  // global
typedef __attribute__((address_space(3))) v4i_t l_v4i;  // LDS
#endif

__device__ __forceinline__ void wait_async_zero() {
#if defined(HAVE_ASYNC_LDS)
#if __has_builtin(__builtin_amdgcn_s_wait_asynccnt)
  __builtin_amdgcn_s_wait_asynccnt(0);
#else
  asm volatile("s_wait_asynccnt 0x0" ::: "memory");
#endif
#endif
}

// ---------------------------------------------------------------------------
// WMMA GEMM: H[M, NCOLS] = X[M, 128] @ W[128, NCOLS]
// One wave computes a 16-row M-tile across all NTILES 16-col N-tiles.
// W staged in LDS once per block (async-to-LDS when no padding is needed).
// A fragments use a depth-4 global prefetch ring to hide load latency
// under the NTILES chained V_WMMA_F32_16X16X4_F32 ops per k-step.
// ---------------------------------------------------------------------------
template <int NCOLS, int NTILES>
__global__ __launch_bounds__(256) void gemm_wmma_f32(
    const float* __restrict__ X,  // [M, 128] row-major
    const float* __restrict__ W,  // [128, NCOLS] row-major
    float* __restrict__ H,        // [M, NCOLS] row-major
    int M) {
  constexpr int NP = NTILES * 16;  // padded column count in LDS
  __shared__ float Wlds[K_DIM * NP];

  if constexpr (NCOLS == NP) {
#if defined(HAVE_ASYNC_LDS)
    // Bulk async stage: 128*NP floats, 16B per lane per op, ASYNCcnt-tracked.
    for (int idx4 = threadIdx.x; idx4 < (K_DIM * NP) / 4; idx4 += 256) {
      __builtin_amdgcn_global_load_async_to_lds_b128(
          (g_v4i*)(W + idx4 * 4), (l_v4i*)(Wlds + idx4 * 4), 0, 0);
    }
    wait_async_zero();
#else
    for (int idx = threadIdx.x; idx < K_DIM * NP; idx += 256)
      Wlds[idx] = W[idx];
#endif
  } else {
    // Zero-padded stage ([k][n] layout, pad n >= NCOLS) for NCOLS=40.
    for (int idx = threadIdx.x; idx < K_DIM * NP; idx += 256) {
      int k = idx / NP, n = idx % NP;
      Wlds[idx] = (n < NCOLS) ? W[k * NCOLS + n] : 0.0f;
    }
  }
  __syncthreads();

  const int wave = threadIdx.x >> 5;
  const int lane = threadIdx.x & 31;
  const int lm   = lane & 15;   // M (for A) / N (for B,C,D) within tile
  const int lhi  = lane >> 4;   // half-wave selector

  const long tile  = (long)blockIdx.x * 8 + wave;  // 16-row tile index
  const long row0  = tile * 16;
  const bool valid = row0 < (long)M;
  const long rbase = valid ? row0 : 0;  // clamp reads; keep EXEC all-ones

  // A fragment source: lane holds row (row0+lm), K = k + 2*lhi + {0,1}
  const float* aptr = X + (rbase + lm) * K_DIM + 2 * lhi;

  v8f acc[NTILES];
#pragma unroll
  for (int n = 0; n < NTILES; ++n)
#pragma unroll
    for (int j = 0; j < 8; ++j) acc[n][j] = 0.0f;

  // Depth-4 prefetch ring over the 32 k-steps.
  constexpr int PF = 4;
  v2f abuf[PF];
#pragma unroll
  for (int i = 0; i < PF; ++i) abuf[i] = *(const v2f*)(aptr + 4 * i);

#pragma unroll
  for (int kk = 0; kk < K_DIM / 4; ++kk) {
    const v2f a = abuf[kk & (PF - 1)];
    if (kk + PF < K_DIM / 4)
      abuf[kk & (PF - 1)] = *(const v2f*)(aptr + 4 * (kk + PF));

    const float* brow = &Wlds[(kk * 4 + 2 * lhi) * NP + lm];
#pragma unroll
    for (int n = 0; n < NTILES; ++n) {
      v2f b;
      b.x = brow[n * 16];       // K = 4*kk + 2*lhi
      b.y = brow[NP + n * 16];  // K = 4*kk + 2*lhi + 1
      acc[n] = __builtin_amdgcn_wmma_f32_16x16x4_f32(
          false, a, false, b, (short)0, acc[n], false, false);
    }
  }

  // C/D layout: lane L, VGPR j -> row = row0 + j + 8*(L/16), col = n*16 + L%16
  if (valid) {
#pragma unroll
    for (int n = 0; n < NTILES; ++n) {
      const int col = n * 16 + lm;
      if (col < NCOLS) {
#pragma unroll
        for (int j = 0; j < 8; ++j) {
          H[(row0 + j + 8 * lhi) * NCOLS + col] = acc[n][j];
        }
      }
    }
  }
}

// ---------------------------------------------------------------------------
// Degree / normalization kernels
// ---------------------------------------------------------------------------
__global__ void fill_ones(float* p, int n) {
  int i = blockIdx.x * blockDim.x + threadIdx.x;
  if (i < n) p[i] = 1.0f;  // self-loop contributes 1 to every node's degree
}

__global__ void deg_accum(const int* __restrict__ dst, float* deg, int E) {
  int e = blockIdx.x * blockDim.x + threadIdx.x;
  if (e < E) atomicAdd(&deg[dst[e]], 1.0f);
}

__global__ void deg_to_dinv(float* d, int n) {
  int i = blockIdx.x * blockDim.x + threadIdx.x;
  if (i < n) {
    float v = d[i];
    d[i] = (v > 0.0f) ? rsqrtf(v) : 0.0f;
  }
}

// ---------------------------------------------------------------------------
// Layer-1 aggregation (128 features)
// ---------------------------------------------------------------------------
__global__ void self_loop_init128(const float* __restrict__ H,
                                  const float* __restrict__ dinv,
                                  float* __restrict__ AGG, int n) {
  long idx = (long)blockIdx.x * blockDim.x + threadIdx.x;  // float4 chunks
  long total = (long)n * 32;
  if (idx >= total) return;
  int node = (int)(idx >> 5);
  float w = dinv[node];
  w *= w;
  float4 v = ((const float4*)H)[idx];
  float4 o = {v.x * w, v.y * w, v.z * w, v.w * w};
  ((float4*)AGG)[idx] = o;
}

__global__ __launch_bounds__(256) void scatter_edges128(
    const int* __restrict__ src, const int* __restrict__ dst,
    const float* __restrict__ dinv, const float* __restrict__ H,
    float* __restrict__ AGG, int E) {
  long gw = ((long)blockIdx.x * blockDim.x + threadIdx.x) >> 5;  // edge id
  int lane = threadIdx.x & 31;
  if (gw >= E) return;
  int s = src[gw], d = dst[gw];
  float nrm = dinv[s] * dinv[d];
  float4 v = ((const float4*)(H + (long)s * 128))[lane];
  float* o = AGG + (long)d * 128 + lane * 4;
  atomicAdd(o + 0, v.x * nrm);
  atomicAdd(o + 1, v.y * nrm);
  atomicAdd(o + 2, v.z * nrm);
  atomicAdd(o + 3, v.w * nrm);
}

__global__ void bias_relu128(float* AGG, const float* __restrict__ b, long total) {
  long idx = (long)blockIdx.x * blockDim.x + threadIdx.x;
  if (idx >= total) return;
  int col = (int)(idx & 127);
  float v = AGG[idx] + b[col];
  AGG[idx] = (v > 0.0f) ? v : 0.0f;
}

// ---------------------------------------------------------------------------
// Layer-2 aggregation (40 features) -> writes directly into d_out
// ---------------------------------------------------------------------------
__global__ void out_init40(const float* __restrict__ H2,
                           const float* __restrict__ dinv,
                           const float* __restrict__ b2,
                           float* __restrict__ OUT, int n) {
  long idx = (long)blockIdx.x * blockDim.x + threadIdx.x;
  long total = (long)n * 40;
  if (idx >= total) return;
  int node = (int)(idx / 40);
  int col  = (int)(idx % 40);
  float w = dinv[node];
  OUT[idx] = H2[idx] * w * w + b2[col];
}

__global__ __launch_bounds__(256) void scatter_edges40(
    const int* __restrict__ src, const int* __restrict__ dst,
    const float* __restrict__ dinv, const float* __restrict__ H2,
    float* __restrict__ OUT, int E) {
  long gw = ((long)blockIdx.x * blockDim.x + threadIdx.x) >> 5;
  int lane = threadIdx.x & 31;
  if (gw >= E) return;
  int s = src[gw], d = dst[gw];
  float nrm = dinv[s] * dinv[d];
  const float* hp = H2 + (long)s * 40;
  float* op = OUT + (long)d * 40;
  for (int c = lane; c < 40; c += 32) {
    atomicAdd(op + c, hp[c] * nrm);
  }
}

// ---------------------------------------------------------------------------
// Launch
// ---------------------------------------------------------------------------
extern "C" void kernel_launch(void* const* d_in, const int* in_sizes, int n_in,
                              void* d_out, int out_size, void* d_ws, size_t ws_size,
                              hipStream_t stream) {
  const float* x  = (const float*)d_in[0];
  const int*   ei = (const int*)d_in[1];
  const float* W1 = (const float*)d_in[2];
  const float* b1 = (const float*)d_in[3];
  const float* W2 = (const float*)d_in[4];
  const float* b2 = (const float*)d_in[5];
  float* out = (float*)d_out;

  const int N = in_sizes[0] / 128;  // 100000 nodes
  const int E = in_sizes[1] / 2;    // 1.6M edges
  const int* src = ei;
  const int* dst = ei + E;

  // Workspace carve-out (256B aligned): dinv | H1 | AGG1 | H2  (~119 MB total)
  char* ws = (char*)d_ws;
  size_t off = 0;
  auto carve = [&](size_t bytes) -> char* {
    char* p = ws + off;
    off += (bytes + 255) & ~(size_t)255;
    return p;
  };
  float* dinv = (float*)carve((size_t)N * 4);
  float* H1   = (float*)carve((size_t)N * 128 * 4);
  float* AGG1 = (float*)carve((size_t)N * 128 * 4);
  float* H2   = (float*)carve((size_t)N * 40 * 4);

  const int B = 256;

  // 1) degree (self-loop baked in) -> dinv (in place)
  fill_ones<<<(N + B - 1) / B, B, 0, stream>>>(dinv, N);
  deg_accum<<<(E + B - 1) / B, B, 0, stream>>>(dst, dinv, E);
  deg_to_dinv<<<(N + B - 1) / B, B, 0, stream>>>(dinv, N);

  // 2) H1 = x @ W1   (WMMA f32)
  const int tiles = (N + 15) / 16;
  gemm_wmma_f32<128, 8><<<(tiles + 7) / 8, 256, 0, stream>>>(x, W1, H1, N);

  // 3) AGG1 = self-loop term; += edge scatter; relu(+b1) in place
  long chunks1 = (long)N * 32;
  self_loop_init128<<<(int)((chunks1 + B - 1) / B), B, 0, stream>>>(H1, dinv, AGG1, N);
  scatter_edges128<<<(int)(((long)E * 32 + B - 1) / B), B, 0, stream>>>(
      src, dst, dinv, H1, AGG1, E);
  long t1 = (long)N * 128;
  bias_relu128<<<(int)((t1 + B - 1) / B), B, 0, stream>>>(AGG1, b1, t1);

  // 4) H2 = AGG1 @ W2  (WMMA f32, N=40 padded to 3 tiles)
  gemm_wmma_f32<40, 3><<<(tiles + 7) / 8, 256, 0, stream>>>(AGG1, W2, H2, N);

  // 5) out = self-loop term + b2; += edge scatter
  long t2 = (long)N * 40;
  out_init40<<<(int)((t2 + B - 1) / B), B, 0, stream>>>(H2, dinv, b2, out, N);
  scatter_edges40<<<(int)(((long)E * 32 + B - 1) / B), B, 0, stream>>>(
      src, dst, dinv, H2, out, E);
}